// AtomNetGraph_13932873908263
// MI455X (gfx1250) — compile-verified
//
#include <hip/hip_runtime.h>

typedef float v2f __attribute__((ext_vector_type(2)));
typedef float v8f __attribute__((ext_vector_type(8)));

#define BN_EPS 1e-5f

// ---------------------------------------------------------------- utilities
__global__ void k_zero(float* __restrict__ p, long long n) {
  long long i = (long long)blockIdx.x * blockDim.x + threadIdx.x;
  long long stride = (long long)gridDim.x * blockDim.x;
  for (; i < n; i += stride) p[i] = 0.0f;
}

__global__ void k_deg(const int* __restrict__ dst, const float* __restrict__ ew,
                      float* __restrict__ deg, int E) {
  int e = blockIdx.x * blockDim.x + threadIdx.x;
  if (e < E) atomicAdd(&deg[dst[e]], ew[e]);
}

__global__ void k_dinv(const float* __restrict__ deg, float* __restrict__ dinv, int n) {
  int i = blockIdx.x * blockDim.x + threadIdx.x;
  if (i < n) dinv[i] = rsqrtf(deg[i] + 1.0f);
}

__global__ void k_norm(const int* __restrict__ src, const int* __restrict__ dst,
                       const float* __restrict__ ew, const float* __restrict__ dinv,
                       float* __restrict__ norm, int E) {
  int e = blockIdx.x * blockDim.x + threadIdx.x;
  if (e < E) norm[e] = dinv[src[e]] * ew[e] * dinv[dst[e]];
}

// ------------------------------------------------------- fp32 WMMA GEMM H=X*W
// One wave32 computes a 16x64 output strip (4 column tiles): each A fragment
// (2 VGPRs) is reused by 4 v_wmma_f32_16x16x4_f32 ops, cutting X re-reads 4x
// and giving 4 independent accumulator chains to hide WMMA latency.
// N is a multiple of 16 and all channel widths are multiples of 64.
__global__ void k_gemm_wmma(const float* __restrict__ X, const float* __restrict__ W,
                            float* __restrict__ H, int nrows, int Cin, int Cout) {
  int wave = (int)((blockIdx.x * (long long)blockDim.x + threadIdx.x) >> 5);
  int lane = threadIdx.x & 31;
  int colGroups = Cout >> 6;                  // 64-column groups
  int nWork = (nrows >> 4) * colGroups;
  if (wave >= nWork) return;                  // wave-uniform: EXEC stays all-1s
  int rowTile = wave / colGroups;
  int colGroup = wave - rowTile * colGroups;
  int half = lane >> 4;                       // K-pair select (K={0,1} vs {2,3})
  int mn   = lane & 15;                       // M for A-frag, N for B-frag

  const float* Arow  = X + (size_t)(rowTile * 16 + mn) * Cin;
  const float* Bbase = W + (size_t)(colGroup * 64 + mn);

  v8f acc0 = {}, acc1 = {}, acc2 = {}, acc3 = {};
  for (int k = 0; k < Cin; k += 4) {
    int kk = k + 2 * half;
    v2f a;
    a.x = Arow[kk];
    a.y = Arow[kk + 1];
    const float* Brow0 = Bbase + (size_t)kk * Cout;
    const float* Brow1 = Brow0 + Cout;
    v2f b0, b1, b2, b3;
    b0.x = Brow0[0];  b0.y = Brow1[0];
    b1.x = Brow0[16]; b1.y = Brow1[16];
    b2.x = Brow0[32]; b2.y = Brow1[32];
    b3.x = Brow0[48]; b3.y = Brow1[48];
    acc0 = __builtin_amdgcn_wmma_f32_16x16x4_f32(false, a, false, b0, (short)0, acc0, false, false);
    acc1 = __builtin_amdgcn_wmma_f32_16x16x4_f32(false, a, false, b1, (short)0, acc1, false, false);
    acc2 = __builtin_amdgcn_wmma_f32_16x16x4_f32(false, a, false, b2, (short)0, acc2, false, false);
    acc3 = __builtin_amdgcn_wmma_f32_16x16x4_f32(false, a, false, b3, (short)0, acc3, false, false);
  }
  // C/D layout: VGPR r -> row (r + 8*half), col = mn (+ tile offset)
  int rbase = rowTile * 16 + 8 * half;
  float* out = H + (size_t)(colGroup * 64 + mn);
#pragma unroll
  for (int r = 0; r < 8; ++r) {
    float* o = out + (size_t)(rbase + r) * Cout;
    o[0]  = acc0[r];
    o[16] = acc1[r];
    o[32] = acc2[r];
    o[48] = acc3[r];
  }
}

// ------------------------------------------------- edge scatter: OUT[dst]+=H[src]*norm
// One thread per (edge, 4-channel chunk). Feature buffers fit in 192MB L2, so
// the random gathers and float atomics land in L2, not HBM.
__global__ void k_scatter(const int* __restrict__ src, const int* __restrict__ dst,
                          const float* __restrict__ norm, const float* __restrict__ H,
                          float* __restrict__ OUT, long long total,
                          int chunkShift, int C) {
  long long t = (long long)blockIdx.x * blockDim.x + threadIdx.x;
  if (t >= total) return;
  int chunks = 1 << chunkShift;
  int e = (int)(t >> chunkShift);
  int q = (int)(t & (chunks - 1)) << 2;
  int s = src[e], d = dst[e];
  float w = norm[e];
  const float4 hv = *(const float4*)(H + (size_t)s * C + q);
  float* o = OUT + (size_t)d * C + q;
  atomicAdd(o + 0, hv.x * w);
  atomicAdd(o + 1, hv.y * w);
  atomicAdd(o + 2, hv.z * w);
  atomicAdd(o + 3, hv.w * w);
}

// ---------------------- self-loop + bias, fused BN statistics (LDS reduction)
__global__ void k_self_stats(const float* __restrict__ H, const float* __restrict__ dinv,
                             const float* __restrict__ bias, float* __restrict__ OUT,
                             float* __restrict__ stats, long long total,
                             int cshift, int C) {
  __shared__ float ssum[256];
  __shared__ float ssq[256];
  int tid = threadIdx.x;
  for (int c = tid; c < C; c += blockDim.x) { ssum[c] = 0.0f; ssq[c] = 0.0f; }
  __syncthreads();

  long long stride = (long long)gridDim.x * blockDim.x;
  for (long long i = (long long)blockIdx.x * blockDim.x + tid; i < total; i += stride) {
    int c = (int)(i & (C - 1));
    int row = (int)(i >> cshift);
    float di = dinv[row];
    float v = OUT[i] + H[i] * di * di + bias[c];
    OUT[i] = v;
    atomicAdd(&ssum[c], v);        // ds_add_f32
    atomicAdd(&ssq[c], v * v);
  }
  __syncthreads();
  for (int c = tid; c < C; c += blockDim.x) {
    atomicAdd(&stats[c], ssum[c]);
    atomicAdd(&stats[C + c], ssq[c]);
  }
}

// -------------------------------------------------- BN apply (+optional ReLU)
__global__ void k_bn(const float* __restrict__ IN, const float* __restrict__ stats,
                     const float* __restrict__ g, const float* __restrict__ bb,
                     float* __restrict__ Y, long long total, int C, int relu,
                     float invN) {
  long long i = (long long)blockIdx.x * blockDim.x + threadIdx.x;
  if (i >= total) return;
  int c = (int)(i & (C - 1));
  float m = stats[c] * invN;
  float var = stats[C + c] * invN - m * m;
  float y = (IN[i] - m) * rsqrtf(var + BN_EPS) * g[c] + bb[c];
  if (relu) y = fmaxf(y, 0.0f);
  Y[i] = y;
}

// ---------------------------------------------------------------- dispatcher
extern "C" void kernel_launch(void* const* d_in, const int* in_sizes, int n_in,
                              void* d_out, int out_size, void* d_ws, size_t ws_size,
                              hipStream_t stream) {
  const float* x  = (const float*)d_in[0];
  const int*   ei = (const int*)d_in[1];
  const float* ew = (const float*)d_in[2];
  const int C_IN = 128;
  const int N = in_sizes[0] / C_IN;
  const int E = in_sizes[2];
  const int* src = ei;
  const int* dst = ei + E;

  const int Cin[5]  = {128, 64, 128, 256, 256};
  const int Cout[5] = {64, 128, 256, 256, 128};
  const int MAXC = 256;

  float* ws   = (float*)d_ws;
  float* Xbuf = ws;  ws += (size_t)N * MAXC;
  float* Hbuf = ws;  ws += (size_t)N * MAXC;
  float* Obuf = ws;  ws += (size_t)N * MAXC;
  float* deg  = ws;  ws += N;
  float* dinv = ws;  ws += N;
  float* norm = ws;  ws += E;
  float* stats = ws; ws += 2 * MAXC;

  // Graph normalization terms: computed once, reused for all 5 layers.
  k_zero<<<256, 256, 0, stream>>>(deg, (long long)N);
  k_deg<<<(E + 255) / 256, 256, 0, stream>>>(dst, ew, deg, E);
  k_dinv<<<(N + 255) / 256, 256, 0, stream>>>(deg, dinv, N);
  k_norm<<<(E + 255) / 256, 256, 0, stream>>>(src, dst, ew, dinv, norm, E);

  const float* Xin = x;
  for (int l = 0; l < 5; ++l) {
    int ci = Cin[l], co = Cout[l];
    const float* W  = (const float*)d_in[3 + 4 * l + 0];
    const float* b  = (const float*)d_in[3 + 4 * l + 1];
    const float* g  = (const float*)d_in[3 + 4 * l + 2];
    const float* bb = (const float*)d_in[3 + 4 * l + 3];
    long long nc = (long long)N * co;
    int cshift = (co == 64) ? 6 : (co == 128) ? 7 : 8;

    k_zero<<<2048, 256, 0, stream>>>(Obuf, nc);
    k_zero<<<1, 256, 0, stream>>>(stats, (long long)(2 * co));

    int nWork = (N / 16) * (co / 64);   // one wave per 16x64 output strip
    k_gemm_wmma<<<(nWork + 7) / 8, 256, 0, stream>>>(Xin, W, Hbuf, N, ci, co);

    int chunkShift = cshift - 2;  // co/4 chunks of float4
    long long totalSc = (long long)E << chunkShift;
    k_scatter<<<(unsigned)((totalSc + 255) / 256), 256, 0, stream>>>(
        src, dst, norm, Hbuf, Obuf, totalSc, chunkShift, co);

    k_self_stats<<<2048, 256, 0, stream>>>(Hbuf, dinv, b, Obuf, stats, nc, cshift, co);

    float* Y = (l == 4) ? (float*)d_out : Xbuf;
    k_bn<<<(unsigned)((nc + 255) / 256), 256, 0, stream>>>(
        Obuf, stats, g, bb, Y, nc, co, (l < 4) ? 1 : 0, 1.0f / (float)N);

    Xin = Xbuf;
  }
}